// GraphLearning_18287970746460
// MI455X (gfx1250) — compile-verified
//
#include <hip/hip_runtime.h>
#include <hip/hip_bf16.h>
#include <math.h>

typedef __attribute__((ext_vector_type(2))) float v2f;
typedef __attribute__((ext_vector_type(8))) float v8f;

#define LDIM 128
#define BATCH 64
#define NEG_SLOPE 0.01f

// ---------------------------------------------------------------------------
// Cooperative 64KB (128x128 f32) global -> LDS stage using the CDNA5 async
// path: per-lane global_load_async_to_lds_b128, tracked by ASYNCcnt.
// 256 threads x 16 iterations x 16B = 64KB.  ASYNCcnt drain + barrier after.
// ---------------------------------------------------------------------------
__device__ __forceinline__ void stage_to_lds(const float* __restrict__ gsrc,
                                             float* lds_dst, int tid) {
    uint32_t lbase = (uint32_t)(size_t)lds_dst;   // LDS aperture: low 32 bits = LDS byte addr
#pragma unroll
    for (int i = tid; i < (LDIM * LDIM) / 4; i += 256) {
        uint64_t ga = (uint64_t)(size_t)(gsrc + i * 4);
        uint32_t la = lbase + (uint32_t)(i * 16);
        asm volatile("global_load_async_to_lds_b128 %0, %1, off"
                     :: "v"(la), "v"(ga) : "memory");
    }
    asm volatile("s_wait_asynccnt 0" ::: "memory");
    __syncthreads();
}

// ---------------------------------------------------------------------------
// Kernel 1: per-batch row squared norms.  grid=B, block=L
// ---------------------------------------------------------------------------
__global__ void k_norms(const float* __restrict__ x, float* __restrict__ norms) {
    int b = blockIdx.x;
    int i = threadIdx.x;
    const float* row = x + ((size_t)b * LDIM + i) * LDIM;
    float s = 0.f;
#pragma unroll 8
    for (int f = 0; f < LDIM; ++f) { float v = row[f]; s += v * v; }
    norms[b * LDIM + i] = s;
}

// ---------------------------------------------------------------------------
// Kernel 2 (WMMA f32 16x16x4): Gram matrix per batch -> adj = dist + I.
// x[b] staged in LDS (async).  Each wave owns one column tile (jt = wave):
// B strip held in registers across all 8 output tiles; A strip bulk-preloaded
// per tile so the 32 WMMAs issue back-to-back.  grid=B, block=256.
// ---------------------------------------------------------------------------
__global__ void k_adj(const float* __restrict__ x, const float* __restrict__ norms,
                      float* __restrict__ adj) {
    __shared__ __align__(16) float Xl[LDIM * LDIM];
    int b   = blockIdx.x;
    int tid = threadIdx.x;
    stage_to_lds(x + (size_t)b * LDIM * LDIM, Xl, tid);

    const float* nr = norms + b * LDIM;
    float* A = adj + (size_t)b * LDIM * LDIM;
    int lane  = tid & 31;
    int wave  = tid >> 5;             // == jt (column tile), fixed per wave
    int r     = lane & 15;            // M (A-frag) / N (B-frag) within tile
    int khalf = (lane >> 4) << 1;     // K sub-offset: 0 (lanes 0-15) or 2 (16-31)

    // Loop-invariant B strip: column tile jt = wave, all K.
    int brow = (wave * 16 + r) * LDIM;
    v2f bfr[32];
#pragma unroll
    for (int kk = 0; kk < 32; ++kk)
        bfr[kk] = *(const v2f*)&Xl[brow + kk * 4 + khalf];

    int n = wave * 16 + r;
    for (int it = 0; it < 8; ++it) {
        int arow = (it * 16 + r) * LDIM;
        v2f af[32];
#pragma unroll
        for (int kk = 0; kk < 32; ++kk)
            af[kk] = *(const v2f*)&Xl[arow + kk * 4 + khalf];

        v8f c = {};
#pragma unroll
        for (int kk = 0; kk < 32; ++kk)
            c = __builtin_amdgcn_wmma_f32_16x16x4_f32(false, af[kk], false, bfr[kk],
                                                      (short)0, c, false, false);

        int mbase = it * 16 + ((lane >> 4) << 3);
#pragma unroll
        for (int v = 0; v < 8; ++v) {
            int m   = mbase + v;
            float g = c[v];
            float sq  = nr[m] + nr[n] - 2.f * g;
            float val = (sq > 0.f) ? sqrtf(sq) : 0.f;   // safe-sqrt semantics
            if (m == n) val += 1.f;                     // + eye(L)
            A[m * LDIM + n] = val;
        }
    }
}

// ---------------------------------------------------------------------------
// Kernel 3: degree + D^-1/2 A D^-1/2, in place on adj.  grid=B, block=L
// ---------------------------------------------------------------------------
__global__ void k_lap(float* __restrict__ adj) {
    int b = blockIdx.x;
    int i = threadIdx.x;
    float* A = adj + (size_t)b * LDIM * LDIM;
    __shared__ float dis[LDIM];
    float deg = 0.f;
#pragma unroll 8
    for (int j = 0; j < LDIM; ++j) deg += A[i * LDIM + j];
    dis[i] = rsqrtf(deg);
    __syncthreads();
    float di = dis[i];
#pragma unroll 8
    for (int j = 0; j < LDIM; ++j) A[i * LDIM + j] = di * dis[j] * A[i * LDIM + j];
}

// ---------------------------------------------------------------------------
// Kernel 4 (WMMA): wa = LeakyReLU(weight @ lap) per batch.
// lap[b] staged in LDS (async).  B strip (lap columns, fixed per wave) hoisted
// to registers once; weight A strips bulk-preloaded per tile from global.
// grid=B, block=256.
// ---------------------------------------------------------------------------
__global__ void k_wa(const float* __restrict__ w, const float* __restrict__ lap,
                     float* __restrict__ wa) {
    __shared__ __align__(16) float Bl[LDIM * LDIM];
    int b   = blockIdx.x;
    int tid = threadIdx.x;
    stage_to_lds(lap + (size_t)b * LDIM * LDIM, Bl, tid);

    float* O = wa + (size_t)b * LDIM * LDIM;
    int lane  = tid & 31;
    int wave  = tid >> 5;             // == jt, fixed per wave
    int r     = lane & 15;
    int khalf = (lane >> 4) << 1;
    int n     = wave * 16 + r;

    // Loop-invariant B strip from LDS: column n of lap, all K.
    v2f bfr[32];
#pragma unroll
    for (int kk = 0; kk < 32; ++kk) {
        int k = kk * 4 + khalf;
        v2f bf;
        bf[0] = Bl[(k + 0) * LDIM + n];
        bf[1] = Bl[(k + 1) * LDIM + n];
        bfr[kk] = bf;
    }

    for (int it = 0; it < 8; ++it) {
        const float* arow = w + (it * 16 + r) * LDIM;
        v2f af[32];
#pragma unroll
        for (int kk = 0; kk < 32; ++kk)
            af[kk] = *(const v2f*)(arow + kk * 4 + khalf);

        v8f c = {};
#pragma unroll
        for (int kk = 0; kk < 32; ++kk)
            c = __builtin_amdgcn_wmma_f32_16x16x4_f32(false, af[kk], false, bfr[kk],
                                                      (short)0, c, false, false);

        int mbase = it * 16 + ((lane >> 4) << 3);
#pragma unroll
        for (int v = 0; v < 8; ++v) {
            float g = c[v];
            g = (g >= 0.f) ? g : NEG_SLOPE * g;         // LeakyReLU
            O[(mbase + v) * LDIM + n] = g;
        }
    }
}

// ---------------------------------------------------------------------------
// Kernel 5: row-wise log_softmax in place.  grid=B, block=L
// ---------------------------------------------------------------------------
__global__ void k_lsm(float* __restrict__ wa) {
    int b = blockIdx.x, i = threadIdx.x;
    float* row = wa + ((size_t)b * LDIM + i) * LDIM;
    float mx = -INFINITY;
#pragma unroll 8
    for (int j = 0; j < LDIM; ++j) mx = fmaxf(mx, row[j]);
    float s = 0.f;
#pragma unroll 8
    for (int j = 0; j < LDIM; ++j) s += expf(row[j] - mx);
    float lse = mx + logf(s);
#pragma unroll 8
    for (int j = 0; j < LDIM; ++j) row[j] -= lse;
}

// ---------------------------------------------------------------------------
// Kernel 6: mean over batch -> m.  grid=64, block=256
// ---------------------------------------------------------------------------
__global__ void k_mean(const float* __restrict__ wa, float* __restrict__ m) {
    int idx = blockIdx.x * blockDim.x + threadIdx.x;   // 0..L*L-1
    float s = 0.f;
#pragma unroll 8
    for (int b = 0; b < BATCH; ++b) s += wa[(size_t)b * LDIM * LDIM + idx];
    m[idx] = s * (1.f / (float)BATCH);
}

// ---------------------------------------------------------------------------
// Kernel 7: out[0] = 0, out[1] = m.  grid=64, block=256
// ---------------------------------------------------------------------------
__global__ void k_out01(const float* __restrict__ m, float* __restrict__ out) {
    int idx = blockIdx.x * blockDim.x + threadIdx.x;
    out[idx] = 0.f;
    out[LDIM * LDIM + idx] = m[idx];
}

// ---------------------------------------------------------------------------
// Kernel 8 (WMMA): D = 2*A@B - C  (C may be null -> treated as zero).
// B staged in LDS (async), B strip hoisted; A strips bulk-preloaded.
// grid=1, block=256.
// ---------------------------------------------------------------------------
__global__ void k_cheb(const float* __restrict__ A, const float* __restrict__ Bm,
                       const float* __restrict__ Cm, float* __restrict__ D) {
    __shared__ __align__(16) float Bl[LDIM * LDIM];
    int tid = threadIdx.x;
    stage_to_lds(Bm, Bl, tid);

    int lane  = tid & 31;
    int wave  = tid >> 5;             // == jt, fixed per wave
    int r     = lane & 15;
    int khalf = (lane >> 4) << 1;
    int n     = wave * 16 + r;

    v2f bfr[32];
#pragma unroll
    for (int kk = 0; kk < 32; ++kk) {
        int k = kk * 4 + khalf;
        v2f bf;
        bf[0] = Bl[(k + 0) * LDIM + n];
        bf[1] = Bl[(k + 1) * LDIM + n];
        bfr[kk] = bf;
    }

    for (int it = 0; it < 8; ++it) {
        const float* arow = A + (it * 16 + r) * LDIM;
        v2f af[32];
#pragma unroll
        for (int kk = 0; kk < 32; ++kk)
            af[kk] = *(const v2f*)(arow + kk * 4 + khalf);

        v8f c = {};
#pragma unroll
        for (int kk = 0; kk < 32; ++kk)
            c = __builtin_amdgcn_wmma_f32_16x16x4_f32(false, af[kk], false, bfr[kk],
                                                      (short)0, c, false, false);

        int mbase = it * 16 + ((lane >> 4) << 3);
#pragma unroll
        for (int v = 0; v < 8; ++v) {
            int m = mbase + v;
            float sub = Cm ? Cm[m * LDIM + n] : 0.f;
            D[m * LDIM + n] = 2.f * c[v] - sub;
        }
    }
}

// ---------------------------------------------------------------------------
extern "C" void kernel_launch(void* const* d_in, const int* in_sizes, int n_in,
                              void* d_out, int out_size, void* d_ws, size_t ws_size,
                              hipStream_t stream) {
    const float* x = (const float*)d_in[0];   // [64,128,128]
    const float* w = (const float*)d_in[1];   // [1,128,128]
    float* out = (float*)d_out;               // [4,128,128]

    // Workspace layout (floats):
    //   norms : B*L            =   8192
    //   adj   : B*L*L          = 1048576  (becomes lap in place)
    //   wa    : B*L*L          = 1048576
    //   m     : L*L            =   16384
    float* ws    = (float*)d_ws;
    float* norms = ws;
    float* adj   = norms + (size_t)BATCH * LDIM;
    float* wa    = adj + (size_t)BATCH * LDIM * LDIM;
    float* m     = wa + (size_t)BATCH * LDIM * LDIM;

    k_norms<<<BATCH, LDIM, 0, stream>>>(x, norms);
    k_adj  <<<BATCH, 256,  0, stream>>>(x, norms, adj);
    k_lap  <<<BATCH, LDIM, 0, stream>>>(adj);
    k_wa   <<<BATCH, 256,  0, stream>>>(w, adj, wa);
    k_lsm  <<<BATCH, LDIM, 0, stream>>>(wa);
    k_mean <<<64,    256,  0, stream>>>(wa, m);
    k_out01<<<64,    256,  0, stream>>>(m, out);

    float* third  = out + 2 * LDIM * LDIM;
    float* fourth = out + 3 * LDIM * LDIM;
    k_cheb<<<1, 256, 0, stream>>>(m, m, nullptr, third);    // third  = 2*m@m
    k_cheb<<<1, 256, 0, stream>>>(m, third, m, fourth);     // fourth = 2*m@third - m
}